// HakesIVF_85753317032256
// MI455X (gfx1250) — compile-verified
//
#include <hip/hip_runtime.h>

// ---------------------------------------------------------------------------
// IVF centroid assignment for MI455X (gfx1250, wave32).
//   d2 = |x|^2 - 2 x.c + |c|^2 ; argmin over c == argmax of (x.c - 0.5|c|^2)
// GEMM runs on V_WMMA_F32_16X16X4_F32 (exact f32, matches reference numerics).
// B fragments are batch-preloaded from LDS and the WMMA accumulation is split
// into two independent chains so LDS latency / accumulator forwarding overlap.
// ---------------------------------------------------------------------------

typedef float v2f __attribute__((ext_vector_type(2)));
typedef float v8f __attribute__((ext_vector_type(8)));

#define DIM        64
#define NLIST      1024
#define CHUNK      128          // centroids staged in LDS per iteration
#define LDS_STRIDE 68           // 64 + 4 pad: conflict-free b64 reads, keeps 16B align
#define WAVES      8            // 256 threads / WG
#define ROWS_PER_WG (16 * WAVES) // 128 rows per workgroup

// Pass 1: |c|^2 per centroid into workspace.
__global__ __launch_bounds__(256) void ivf_csq_kernel(
    const float* __restrict__ cents, float* __restrict__ csq) {
  int c = blockIdx.x * blockDim.x + threadIdx.x;
  if (c < NLIST) {
    const float* row = cents + (long)c * DIM;
    float s = 0.f;
#pragma unroll
    for (int k = 0; k < DIM; k += 4) {
      float4 v = *(const float4*)(row + k);
      s += v.x * v.x + v.y * v.y + v.z * v.z + v.w * v.w;
    }
    csq[c] = s;
  }
}

// Pass 2: WMMA scoring + argmax + gather.
__global__ __launch_bounds__(256) void ivf_assign_kernel(
    const float* __restrict__ vecs, const float* __restrict__ cents,
    const float* __restrict__ csq, float* __restrict__ out) {
  __shared__ float lds[CHUNK * LDS_STRIDE];

  const int tid  = threadIdx.x;
  const int wave = tid >> 5;
  const int lane = tid & 31;
  const int c16  = lane & 15;   // column / row-within-tile selector
  const int hi   = lane >> 4;   // half-wave: selects K pair (A/B) and M group (C)

  const long rbase = (long)blockIdx.x * ROWS_PER_WG + wave * 16;

  // --- Preload A fragments (16x4 f32 layout: lane%16 = M, vgpr+2*hi = K) ---
  v2f a[16];
  {
    const float* vrow = vecs + (rbase + c16) * DIM + 2 * hi;
#pragma unroll
    for (int s = 0; s < 16; ++s)
      a[s] = *(const v2f*)(vrow + 4 * s);
  }

  float best_score[8];
  int   best_idx[8];
#pragma unroll
  for (int j = 0; j < 8; ++j) { best_score[j] = -3.0e38f; best_idx[j] = 0; }

  for (int chunk = 0; chunk < NLIST / CHUNK; ++chunk) {
    __syncthreads();
    // Stage CHUNK centroids (32 KB) into padded LDS, float4 granularity.
#pragma unroll
    for (int it = 0; it < (CHUNK * DIM / 4) / 256; ++it) {  // 8 iters
      int slot = it * 256 + tid;
      int c    = slot >> 4;        // centroid within chunk
      int k4   = slot & 15;        // float4 within row
      float4 v = *(const float4*)(cents + (long)(chunk * CHUNK + c) * DIM + k4 * 4);
      *(float4*)(&lds[c * LDS_STRIDE + k4 * 4]) = v;
    }
    __syncthreads();

    for (int cb = 0; cb < CHUNK / 16; ++cb) {
      const int cg = chunk * CHUNK + cb * 16 + c16;  // this lane's column centroid
      const float half_csq = 0.5f * csq[cg];

      // Batch-preload all B fragments for this 16x64 centroid block so the
      // ds_loads are issued up front and overlap with the WMMA chain.
      v2f b[16];
      const float* brow = &lds[(cb * 16 + c16) * LDS_STRIDE + 2 * hi];
#pragma unroll
      for (int s = 0; s < 16; ++s)
        b[s] = *(const v2f*)(brow + 4 * s);

      // Two independent accumulation chains (even/odd K-steps) to keep the
      // matrix pipe busy despite the serial C dependency.
      v8f acc0 = {0.f, 0.f, 0.f, 0.f, 0.f, 0.f, 0.f, 0.f};
      v8f acc1 = {0.f, 0.f, 0.f, 0.f, 0.f, 0.f, 0.f, 0.f};
#pragma unroll
      for (int s = 0; s < 16; s += 2) {
        acc0 = __builtin_amdgcn_wmma_f32_16x16x4_f32(
            false, a[s],     false, b[s],     (short)0, acc0, false, false);
        acc1 = __builtin_amdgcn_wmma_f32_16x16x4_f32(
            false, a[s + 1], false, b[s + 1], (short)0, acc1, false, false);
      }

#pragma unroll
      for (int j = 0; j < 8; ++j) {
        float sc = (acc0[j] + acc1[j]) - half_csq;  // row m = j + 8*hi, centroid cg
        if (sc > best_score[j]) { best_score[j] = sc; best_idx[j] = cg; }
      }
    }
  }

  // --- Argmax reduction across the 16 lanes sharing each row (masks 1..8
  //     stay inside a half-wave on wave32). Ties -> smaller index. ---
#pragma unroll
  for (int j = 0; j < 8; ++j) {
    float s = best_score[j];
    int   i = best_idx[j];
#pragma unroll
    for (int m = 1; m <= 8; m <<= 1) {
      float os = __shfl_xor(s, m, 32);
      int   oi = __shfl_xor(i, m, 32);
      if (os > s || (os == s && oi < i)) { s = os; i = oi; }
    }
    best_idx[j] = i;
  }

  // --- Gather winning centroid rows (L2-hot) to output. Each half-wave
  //     writes one 256B row per j with float4s. ---
#pragma unroll
  for (int j = 0; j < 8; ++j) {
    long row = rbase + j + 8 * hi;
    long idx = best_idx[j];
    float4 v = *(const float4*)(cents + idx * DIM + c16 * 4);
    *(float4*)(out + row * DIM + c16 * 4) = v;
  }
}

extern "C" void kernel_launch(void* const* d_in, const int* in_sizes, int n_in,
                              void* d_out, int out_size, void* d_ws, size_t ws_size,
                              hipStream_t stream) {
  const float* vecs  = (const float*)d_in[0];   // (262144, 64) f32
  const float* cents = (const float*)d_in[1];   // (1024, 64) f32
  float* out = (float*)d_out;                   // (262144, 64) f32
  float* csq = (float*)d_ws;                    // 1024 f32 scratch

  const int N = in_sizes[0] / DIM;              // 262144

  ivf_csq_kernel<<<(NLIST + 255) / 256, 256, 0, stream>>>(cents, csq);
  ivf_assign_kernel<<<N / ROWS_PER_WG, 256, 0, stream>>>(vecs, cents, csq, out);
}